// SurfaceNormals_new_50809463111741
// MI455X (gfx1250) — compile-verified
//
#include <hip/hip_runtime.h>

#define IMG_H 1024
#define IMG_W 1024
#define IMG_B 8

constexpr int TW = 32;          // tile width  (output pixels)
constexpr int THT = 32;         // tile height (output pixels)
constexpr int PW = TW + 2;      // 34 (halo)
constexpr int PH = THT + 2;     // 34 (halo)
constexpr int PN = PW * PH;     // 1156 tile elements incl. halo
constexpr int NTHREADS = 256;   // 32 x 8
constexpr int NITER = (PN + NTHREADS - 1) / NTHREADS;  // 5

__global__ __launch_bounds__(NTHREADS)
void surface_normals_kernel(const float* __restrict__ depth,
                            const float* __restrict__ Kmat,
                            const int*   __restrict__ crop,
                            float* __restrict__ out)
{
    __shared__ float dtile[PN];          // staged depth tile (async global->LDS)
    __shared__ float Xs[PN];             // point cloud x
    __shared__ float Ys[PN];             // point cloud y
    __shared__ float Zs[PN];             // point cloud z
    __shared__ float sKi[9];             // Kinv for this batch
    __shared__ float sUV[2];             // u/v crop offsets (+0.5 shift folded in)

    const int tx    = threadIdx.x;               // 0..31
    const int ty    = threadIdx.y;               // 0..7
    const int tid   = ty * 32 + tx;
    const int tileX = blockIdx.x * TW;
    const int tileY = blockIdx.y * THT;
    const int b     = blockIdx.z;

    // ---- one lane: general 3x3 inverse of K (adjugate / det) + crop offsets ----
    if (tid == 0) {
        const float* Kb = Kmat + b * 9;
        float a = Kb[0], bb = Kb[1], c = Kb[2];
        float d = Kb[3], e  = Kb[4], f = Kb[5];
        float g = Kb[6], h  = Kb[7], i = Kb[8];
        float det = a * (e * i - f * h) - bb * (d * i - f * g) + c * (d * h - e * g);
        float id = 1.0f / det;
        sKi[0] = (e * i - f * h) * id;  sKi[1] = (c * h - bb * i) * id;  sKi[2] = (bb * f - c * e) * id;
        sKi[3] = (f * g - d * i) * id;  sKi[4] = (a * i - c * g) * id;   sKi[5] = (c * d - a * f) * id;
        sKi[6] = (d * h - e * g) * id;  sKi[7] = (bb * g - a * h) * id;  sKi[8] = (a * e - bb * d) * id;
        sUV[0] = (float)crop[b * 4 + 2] + 0.5f;  // u offset (w crop + shift)
        sUV[1] = (float)crop[b * 4 + 0] + 0.5f;  // v offset (h crop + shift)
    }

    // ---- phase 1: async-copy depth tile (clamped halo) straight into LDS ----
    {
        unsigned long long src =
            (unsigned long long)(const void*)(depth + (size_t)b * IMG_H * IMG_W);
        #pragma unroll
        for (int it = 0; it < NITER; ++it) {
            int t = it * NTHREADS + tid;
            t = (t < PN) ? t : (PN - 1);             // keep all lanes active (dup write benign)
            int ly = t / PW, lx = t - ly * PW;
            int gy = tileY + ly - 1;  gy = gy < 0 ? 0 : (gy > IMG_H - 1 ? IMG_H - 1 : gy);
            int gx = tileX + lx - 1;  gx = gx < 0 ? 0 : (gx > IMG_W - 1 ? IMG_W - 1 : gx);
            unsigned int goff   = (unsigned int)((gy * IMG_W + gx) * 4);
            unsigned int ldsoff = (unsigned int)(unsigned long long)(&dtile[t]);
            // CDNA5 async global->LDS (ASYNCcnt-tracked, no VGPR data round-trip)
            asm volatile("global_load_async_to_lds_b32 %0, %1, %2"
                         :: "v"(ldsoff), "v"(goff), "s"(src)
                         : "memory");
        }
        asm volatile("s_wait_asynccnt 0" ::: "memory");
    }
    __syncthreads();

    // ---- phase 2: unproject every halo-tile element once; share pts via LDS ----
    const float ki00 = sKi[0], ki01 = sKi[1], ki02 = sKi[2];
    const float ki10 = sKi[3], ki11 = sKi[4], ki12 = sKi[5];
    const float ki20 = sKi[6], ki21 = sKi[7], ki22 = sKi[8];
    const float uoff = sUV[0], voff = sUV[1];

    #pragma unroll
    for (int it = 0; it < NITER; ++it) {
        int t = it * NTHREADS + tid;
        t = (t < PN) ? t : (PN - 1);
        int ly = t / PW, lx = t - ly * PW;
        int gy = tileY + ly - 1;  gy = gy < 0 ? 0 : (gy > IMG_H - 1 ? IMG_H - 1 : gy);
        int gx = tileX + lx - 1;  gx = gx < 0 ? 0 : (gx > IMG_W - 1 ? IMG_W - 1 : gx);
        float u  = uoff + (float)gx;
        float v  = voff + (float)gy;
        float p0 = ki00 * u + ki01 * v + ki02;
        float p1 = ki10 * u + ki11 * v + ki12;
        float p2 = ki20 * u + ki21 * v + ki22;
        float rp2 = 1.0f / p2;                       // pts /= pts.z
        float dd  = (dtile[t] + 1.0f) * 0.5f;        // depth remap
        Xs[t] = p0 * rp2 * dd;
        Ys[t] = p1 * rp2 * dd;
        Zs[t] = dd;
    }
    __syncthreads();

    // ---- phase 3: np.gradient stencil + cross product + normalize ----
    const int gx = tileX + tx;
    const float su = (gx > 0 && gx < IMG_W - 1) ? 0.5f : 1.0f;
    const size_t plane = (size_t)IMG_H * IMG_W;

    #pragma unroll
    for (int r = 0; r < THT / 8; ++r) {
        int lyi = ty + r * 8;                 // 0..31 within tile
        int gy  = tileY + lyi;
        const float sv = (gy > 0 && gy < IMG_H - 1) ? 0.5f : 1.0f;
        int c0 = (lyi + 1) * PW + (tx + 1);   // center index in padded tile

        float dxdu = (Xs[c0 + 1]  - Xs[c0 - 1])  * su;
        float dydu = (Ys[c0 + 1]  - Ys[c0 - 1])  * su;
        float dzdu = (Zs[c0 + 1]  - Zs[c0 - 1])  * su;
        float dxdv = (Xs[c0 + PW] - Xs[c0 - PW]) * sv;
        float dydv = (Ys[c0 + PW] - Ys[c0 - PW]) * sv;
        float dzdv = (Zs[c0 + PW] - Zs[c0 - PW]) * sv;

        float nx = dydv * dzdu - dydu * dzdv;
        float ny = dzdv * dxdu - dzdu * dxdv;
        float nz = dxdv * dydu - dxdu * dydv;

        float nn  = sqrtf(nx * nx + ny * ny + nz * nz);
        float inv = 1.0f / fmaxf(nn, 1e-12f);

        // Output is write-once / never re-read: stream it past L2 (TH_STORE_NT)
        // so depth stays resident for inter-tile halo reuse.
        size_t base = (size_t)b * 3 * plane + (size_t)gy * IMG_W + (size_t)gx;
        __builtin_nontemporal_store(nx * inv, &out[base]);
        __builtin_nontemporal_store(ny * inv, &out[base + plane]);
        __builtin_nontemporal_store(nz * inv, &out[base + 2 * plane]);
    }
}

extern "C" void kernel_launch(void* const* d_in, const int* in_sizes, int n_in,
                              void* d_out, int out_size, void* d_ws, size_t ws_size,
                              hipStream_t stream) {
    (void)in_sizes; (void)n_in; (void)out_size; (void)d_ws; (void)ws_size;
    const float* depth = (const float*)d_in[0];
    const float* K     = (const float*)d_in[1];
    const int*   crop  = (const int*)d_in[2];
    float*       out   = (float*)d_out;

    dim3 grid(IMG_W / TW, IMG_H / THT, IMG_B);
    dim3 block(32, 8, 1);
    surface_normals_kernel<<<grid, block, 0, stream>>>(depth, K, crop, out);
}